// DeformableTransformer_52510270161222
// MI455X (gfx1250) — compile-verified
//
#include <hip/hip_runtime.h>

typedef __attribute__((ext_vector_type(2))) float v2f;
typedef __attribute__((ext_vector_type(8))) float v8f;

#define S_TOT  7681
#define S_PAD  7696      // 481 * 16
#define DMODEL 256
#define NHEAD  8
#define HDIM   32
#define NLAY   3
#define DFFN   1024

__device__ __constant__ int c_H[4]  = {76, 38, 19, 10};
__device__ __constant__ int c_W[4]  = {76, 38, 19, 10};
__device__ __constant__ int c_ST[4] = {0, 5776, 7220, 7581};

// ---------------------------------------------------------------------------
// Copy src into padded workspace buffer, zero the pad rows.
// ---------------------------------------------------------------------------
__global__ void init_pad_kernel(const float* __restrict__ src, float* __restrict__ dst) {
    int idx = blockIdx.x * blockDim.x + threadIdx.x;
    int total = S_PAD * DMODEL;
    if (idx >= total) return;
    dst[idx] = (idx < S_TOT * DMODEL) ? src[idx] : 0.0f;
}

// q = out + pos (valid rows only)
__global__ void add_kernel(const float* __restrict__ a, const float* __restrict__ b,
                           float* __restrict__ c) {
    int idx = blockIdx.x * blockDim.x + threadIdx.x;
    if (idx >= S_TOT * DMODEL) return;
    c[idx] = a[idx] + b[idx];
}

// ---------------------------------------------------------------------------
// Repack B (K x N row-major) into K-pair-interleaved layout so each WMMA
// B fragment is one contiguous float2 per lane:
//   BT[((k>>1)*N + n)*2 + (k&1)] = B[k*N + n]
// ---------------------------------------------------------------------------
__global__ void pack_b_kernel(const float* __restrict__ B, float* __restrict__ BT,
                              int N, int K) {
    int idx = blockIdx.x * blockDim.x + threadIdx.x;
    if (idx >= N * K) return;
    int k = idx / N;
    int n = idx - k * N;
    BT[((size_t)(k >> 1) * N + n) * 2 + (k & 1)] = B[idx];
}

// ---------------------------------------------------------------------------
// f32 WMMA GEMM:  D[M,N] = A[M,K] @ B[K,N] + bias  (optional ReLU)
// B is pre-packed by pack_b_kernel (v2f per lane per fragment).
// One wave per 16x64 output strip: 4 independent 16x16 f32 accumulators.
// Each K-step: 1 b64 A load + 4 b64 B loads -> 4 back-to-back WMMAs
// (register-level A reuse; no D->A/B RAW hazard between accumulators).
// A lane layout (16x4, MxK): lanes 0-15 -> M=lane, v0=K0,v1=K1;
//                            lanes 16-31 -> M=lane-16, v0=K2,v1=K3.
// B lane layout (4x16, KxN): mirrored (row K striped across lanes per VGPR).
// C/D layout: VGPR r -> row r + 8*(lane>=16), col = lane&15.
// Requires: Mr % 16 == 0, N % 64 == 0, K % 4 == 0.
// ---------------------------------------------------------------------------
__global__ void __launch_bounds__(128)
gemm_wmma_f32(const float* __restrict__ A, const float* __restrict__ BT,
              const float* __restrict__ bias, float* __restrict__ D,
              int Mr, int N, int K, int relu) {
    const int lane = threadIdx.x & 31;
    const int wave = blockIdx.x * (blockDim.x >> 5) + (threadIdx.x >> 5);
    const int nStrips = N >> 6;                    // 64-wide strips
    const int strips  = (Mr >> 4) * nStrips;
    if (wave >= strips) return;
    const int rowT = wave / nStrips;
    const int colS = (wave - rowT * nStrips) << 6; // strip base column
    const int mi   = lane & 15;
    const int khi  = lane >> 4;

    const float* aptr = A + (size_t)(rowT * 16 + mi) * K + 2 * khi;
    // v2f element j*N + n holds (B[2j][n], B[2j+1][n]); we need j = k/2 + khi.
    const v2f* bptr = (const v2f*)BT + (size_t)khi * N + colS + mi;

    v8f acc0 = {}, acc1 = {}, acc2 = {}, acc3 = {};
    for (int k = 0; k < K; k += 4) {
        const v2f a = *(const v2f*)(aptr + k);
        const v2f* brow = bptr + (size_t)(k >> 1) * N;
        const v2f b0 = brow[0];
        const v2f b1 = brow[16];
        const v2f b2 = brow[32];
        const v2f b3 = brow[48];
        acc0 = __builtin_amdgcn_wmma_f32_16x16x4_f32(false, a, false, b0, (short)0, acc0, false, false);
        acc1 = __builtin_amdgcn_wmma_f32_16x16x4_f32(false, a, false, b1, (short)0, acc1, false, false);
        acc2 = __builtin_amdgcn_wmma_f32_16x16x4_f32(false, a, false, b2, (short)0, acc2, false, false);
        acc3 = __builtin_amdgcn_wmma_f32_16x16x4_f32(false, a, false, b3, (short)0, acc3, false, false);
    }

    const int row0 = rowT * 16 + 8 * khi;
#define GEMM_EPILOGUE(T, ACC)                                                  \
    {                                                                          \
        const int n = colS + (T)*16 + mi;                                      \
        const float bv = bias ? bias[n] : 0.0f;                                \
        _Pragma("unroll") for (int r = 0; r < 8; ++r) {                        \
            float v = (ACC)[r] + bv;                                           \
            if (relu) v = fmaxf(v, 0.0f);                                      \
            D[(size_t)(row0 + r) * N + n] = v;                                 \
        }                                                                      \
    }
    GEMM_EPILOGUE(0, acc0)
    GEMM_EPILOGUE(1, acc1)
    GEMM_EPILOGUE(2, acc2)
    GEMM_EPILOGUE(3, acc3)
#undef GEMM_EPILOGUE
}

// ---------------------------------------------------------------------------
// Multi-scale deformable attention sampling.
// One wave32 per (token s, head m); lane == channel within head (HDIM==32).
// ---------------------------------------------------------------------------
__global__ void __launch_bounds__(256)
msda_kernel(const float* __restrict__ val, const float* __restrict__ off,
            const float* __restrict__ att, const float* __restrict__ vr,
            float* __restrict__ outp) {
    const int lane = threadIdx.x & 31;
    const int wid  = blockIdx.x * (blockDim.x >> 5) + (threadIdx.x >> 5);
    const int s    = wid >> 3;      // / NHEAD
    const int mh   = wid & 7;
    if (s >= S_TOT) return;

    // token's own level -> base reference point
    const int lvl0 = (s < 5776) ? 0 : (s < 7220) ? 1 : (s < 7581) ? 2 : 3;
    const int rel = s - c_ST[lvl0];
    const int W0 = c_W[lvl0];
    const int iy = rel / W0;
    const int ix = rel - iy * W0;
    const float bx = (ix + 0.5f) / (vr[lvl0 * 2 + 0] * (float)c_W[lvl0]);
    const float by = (iy + 0.5f) / (vr[lvl0 * 2 + 1] * (float)c_H[lvl0]);

    // softmax over the 16 (level,point) logits of this head
    float logit[16];
    float mx = -3.4e38f;
#pragma unroll
    for (int j = 0; j < 16; ++j) {
        logit[j] = att[(size_t)s * 128 + mh * 16 + j];
        mx = fmaxf(mx, logit[j]);
    }
    float ssum = 0.0f;
#pragma unroll
    for (int j = 0; j < 16; ++j) { logit[j] = __expf(logit[j] - mx); ssum += logit[j]; }
    const float inv = 1.0f / ssum;

    float accv = 0.0f;
    for (int lvl = 0; lvl < 4; ++lvl) {
        const int W = c_W[lvl], H = c_H[lvl], st = c_ST[lvl];
        const float rx = bx * vr[lvl * 2 + 0];
        const float ry = by * vr[lvl * 2 + 1];
        const float* vbase = val + (size_t)st * DMODEL + mh * HDIM + lane;
        for (int p = 0; p < 4; ++p) {
            const size_t oidx = (size_t)s * DMODEL + mh * 32 + lvl * 8 + p * 2;
            const float lx = rx + off[oidx + 0] / (float)W;
            const float ly = ry + off[oidx + 1] / (float)H;
            const float x = lx * (float)W - 0.5f;
            const float y = ly * (float)H - 0.5f;
            const float x0f = floorf(x), y0f = floorf(y);
            const float dx = x - x0f, dy = y - y0f;
            const int x0 = (int)x0f, y0 = (int)y0f;
            const float w00 = (1.f - dy) * (1.f - dx), w01 = (1.f - dy) * dx;
            const float w10 = dy * (1.f - dx),         w11 = dy * dx;
            float sv = 0.0f;
            if (y0 >= 0 && y0 < H) {
                if (x0 >= 0 && x0 < W)
                    sv += w00 * vbase[(size_t)(y0 * W + x0) * DMODEL];
                if (x0 + 1 >= 0 && x0 + 1 < W)
                    sv += w01 * vbase[(size_t)(y0 * W + x0 + 1) * DMODEL];
            }
            if (y0 + 1 >= 0 && y0 + 1 < H) {
                if (x0 >= 0 && x0 < W)
                    sv += w10 * vbase[(size_t)((y0 + 1) * W + x0) * DMODEL];
                if (x0 + 1 >= 0 && x0 + 1 < W)
                    sv += w11 * vbase[(size_t)((y0 + 1) * W + x0 + 1) * DMODEL];
            }
            accv += (logit[lvl * 4 + p] * inv) * sv;
        }
    }
    outp[(size_t)s * DMODEL + mh * HDIM + lane] = accv;
}

// ---------------------------------------------------------------------------
// out = LayerNorm(out + add) with per-channel gamma/beta. One block per token.
// ---------------------------------------------------------------------------
__global__ void __launch_bounds__(256)
add_ln_kernel(float* __restrict__ out, const float* __restrict__ add,
              const float* __restrict__ g, const float* __restrict__ b) {
    __shared__ float red[DMODEL];
    const int s = blockIdx.x;
    const int t = threadIdx.x;
    const float x = out[(size_t)s * DMODEL + t] + add[(size_t)s * DMODEL + t];

    red[t] = x;
    __syncthreads();
    for (int o = DMODEL / 2; o > 0; o >>= 1) {
        if (t < o) red[t] += red[t + o];
        __syncthreads();
    }
    const float mu = red[0] * (1.0f / DMODEL);
    __syncthreads();

    const float d = x - mu;
    red[t] = d * d;
    __syncthreads();
    for (int o = DMODEL / 2; o > 0; o >>= 1) {
        if (t < o) red[t] += red[t + o];
        __syncthreads();
    }
    const float var = red[0] * (1.0f / DMODEL);
    out[(size_t)s * DMODEL + t] = d * rsqrtf(var + 1e-5f) * g[t] + b[t];
}

// ---------------------------------------------------------------------------
static inline void launch_gemm(const float* A, const float* B, float* BT,
                               const float* bias, float* D, int N, int K,
                               int relu, hipStream_t st) {
    // repack B once (amortized over all 481 row tiles)
    pack_b_kernel<<<(N * K + 255) / 256, 256, 0, st>>>(B, BT, N, K);
    const int strips = (S_PAD >> 4) * (N >> 6);    // 16x64 strips per wave
    const int blocks = (strips + 3) / 4;           // 4 waves per 128-thread block
    gemm_wmma_f32<<<blocks, 128, 0, st>>>(A, BT, bias, D, S_PAD, N, K, relu);
}

extern "C" void kernel_launch(void* const* d_in, const int* in_sizes, int n_in,
                              void* d_out, int out_size, void* d_ws, size_t ws_size,
                              hipStream_t stream) {
    const float* src    = (const float*)d_in[0];
    const float* pos    = (const float*)d_in[1];
    const float* vr     = (const float*)d_in[2];
    const float* W_off  = (const float*)d_in[6];
    const float* b_off  = (const float*)d_in[7];
    const float* W_attn = (const float*)d_in[8];
    const float* b_attn = (const float*)d_in[9];
    const float* W_val  = (const float*)d_in[10];
    const float* b_val  = (const float*)d_in[11];
    const float* W_out  = (const float*)d_in[12];
    const float* b_out  = (const float*)d_in[13];
    const float* ln1_g  = (const float*)d_in[14];
    const float* ln1_b  = (const float*)d_in[15];
    const float* W_ff1  = (const float*)d_in[16];
    const float* b_ff1  = (const float*)d_in[17];
    const float* W_ff2  = (const float*)d_in[18];
    const float* b_ff2  = (const float*)d_in[19];
    const float* ln2_g  = (const float*)d_in[20];
    const float* ln2_b  = (const float*)d_in[21];

    float* ws  = (float*)d_ws;
    float* OUT = ws;                                  // S_PAD x 256
    float* Q   = OUT + (size_t)S_PAD * DMODEL;        // S_PAD x 256 (also TMP)
    float* VAL = Q   + (size_t)S_PAD * DMODEL;        // S_PAD x 256
    float* OFF = VAL + (size_t)S_PAD * DMODEL;        // S_PAD x 256
    float* ATT = OFF + (size_t)S_PAD * DMODEL;        // S_PAD x 128
    float* MS  = ATT + (size_t)S_PAD * 128;           // S_PAD x 256
    float* HB  = MS  + (size_t)S_PAD * DMODEL;        // S_PAD x 1024
    float* BT  = HB  + (size_t)S_PAD * DFFN;          // packed-B scratch (<=256K f)
    float* TMP = Q;                                   // reuse (Q dead after ATT gemm)

    {
        int total = S_PAD * DMODEL;
        init_pad_kernel<<<(total + 255) / 256, 256, 0, stream>>>(src, OUT);
    }

    const int nEl = S_TOT * DMODEL;
    for (int l = 0; l < NLAY; ++l) {
        const float* Wv = W_val  + (size_t)l * DMODEL * DMODEL;
        const float* Wo = W_off  + (size_t)l * DMODEL * 256;
        const float* Wa = W_attn + (size_t)l * DMODEL * 128;
        const float* Wp = W_out  + (size_t)l * DMODEL * DMODEL;
        const float* W1 = W_ff1  + (size_t)l * DMODEL * DFFN;
        const float* W2 = W_ff2  + (size_t)l * DFFN * DMODEL;

        // q = out + pos
        add_kernel<<<(nEl + 255) / 256, 256, 0, stream>>>(OUT, pos, Q);
        // value = out @ Wv + bv
        launch_gemm(OUT, Wv, BT, b_val + l * DMODEL, VAL, DMODEL, DMODEL, 0, stream);
        // sampling offsets / attention logits
        launch_gemm(Q, Wo, BT, b_off + l * 256, OFF, 256, DMODEL, 0, stream);
        launch_gemm(Q, Wa, BT, b_attn + l * 128, ATT, 128, DMODEL, 0, stream);
        // deformable sampling -> MS
        msda_kernel<<<S_TOT, 256, 0, stream>>>(VAL, OFF, ATT, vr, MS);
        // output projection
        launch_gemm(MS, Wp, BT, b_out + l * DMODEL, TMP, DMODEL, DMODEL, 0, stream);
        // out = LN(out + attn)
        add_ln_kernel<<<S_TOT, DMODEL, 0, stream>>>(OUT, TMP, ln1_g + l * DMODEL,
                                                    ln1_b + l * DMODEL);
        // FFN
        launch_gemm(OUT, W1, BT, b_ff1 + l * DFFN, HB, DFFN, DMODEL, 1, stream);
        launch_gemm(HB, W2, BT, b_ff2 + l * DMODEL, TMP, DMODEL, DFFN, 0, stream);
        add_ln_kernel<<<S_TOT, DMODEL, 0, stream>>>(OUT, TMP, ln2_g + l * DMODEL,
                                                    ln2_b + l * DMODEL);
    }

    hipMemcpyAsync(d_out, OUT, (size_t)S_TOT * DMODEL * sizeof(float),
                   hipMemcpyDeviceToDevice, stream);
}